// kNN_layer_26620207301312
// MI455X (gfx1250) — compile-verified
//
#include <hip/hip_runtime.h>
#include <hip/hip_bf16.h>
#include <math.h>

// ---------------- configuration ----------------
#define KD        128          // feature dim (reference: D = 128)
#define MT        32           // query rows per workgroup
#define WAVES     4            // waves per workgroup (blockDim = 128, wave32)
#define SLICE     2000         // train columns per N-slice (125 chunks of 16)
#define TOPK      32           // kpoints (reference scalar == 32)
#define QP        132          // LDS pitch for query tile (conflict-free)
#define SENT      3.0e38f      // +inf sentinel

typedef __attribute__((ext_vector_type(2))) float v2f;
typedef __attribute__((ext_vector_type(8))) float v8f;

// ---------------- kernel 0: row norms ----------------
__global__ void knn_norms(const float* __restrict__ q, const float* __restrict__ t,
                          float* __restrict__ tn, float* __restrict__ qn,
                          int B, int N, int D) {
    int i = blockIdx.x * blockDim.x + threadIdx.x;
    if (i < N) {
        const float* r = t + (size_t)i * D;
        float s = 0.f;
        for (int d = 0; d < D; d += 4) {
            float4 v = *(const float4*)(r + d);
            s += v.x * v.x + v.y * v.y + v.z * v.z + v.w * v.w;
        }
        tn[i] = s;
    } else if (i < N + B) {
        int j = i - N;
        const float* r = q + (size_t)j * D;
        float s = 0.f;
        for (int d = 0; d < D; d += 4) {
            float4 v = *(const float4*)(r + d);
            s += v.x * v.x + v.y * v.y + v.z * v.z + v.w * v.w;
        }
        qn[j] = s;
    }
}

// ---------------- kernel 1: fused WMMA distance + streaming top-k ----------------
// grid.x = ceil(B/32) query tiles, grid.y = NS slices of SLICE train points.
// Wave w processes column-chunks w, w+4, ... of its slice.
// score(q,c) = ||t_c||^2 - 2 q.t_c  (query norm added later; row-constant)
__global__ __launch_bounds__(WAVES * 32)
void knn_phase1(const float* __restrict__ q, const float* __restrict__ t,
                const float* __restrict__ tn,
                float* __restrict__ candV, int* __restrict__ candI,
                int B, int N, int NS) {
    __shared__ float qlds[MT * QP];              // -2 * query tile
    __shared__ float sc[WAVES][32 * 17];         // per-wave 32x16 score tile, pitch 17
    __shared__ float topv[WAVES][32][33];        // per-wave per-row top-k values
    __shared__ int   topi[WAVES][32][33];        // per-wave per-row top-k indices

    const int tid = threadIdx.x;
    const int wv  = tid >> 5;
    const int ln  = tid & 31;
    const int hi  = ln >> 4;        // half-wave select
    const int lc  = ln & 15;        // column within chunk / row within A half
    const int qbase = blockIdx.x * MT;
    const int sbase = blockIdx.y * SLICE;

    // stage -2*queries into LDS (row-clamped for tail tiles)
    for (int i = tid; i < MT * KD; i += WAVES * 32) {
        int r = i >> 7, c = i & (KD - 1);
        int qr = qbase + r; if (qr > B - 1) qr = B - 1;
        qlds[r * QP + c] = -2.0f * q[(size_t)qr * KD + c];
    }
    // init per-lane top-k buffer (lane ln owns query row ln of this wave)
    for (int s = 0; s < TOPK; ++s) { topv[wv][ln][s] = SENT; topi[wv][ln][s] = 0; }
    __syncthreads();

    float curMax = SENT;
    const int cols = (N - sbase < SLICE) ? (N - sbase) : SLICE;
    const int nchunks = (cols + 15) >> 4;

    for (int ch = wv; ch < nchunks; ch += WAVES) {
        const int cbase = sbase + ch * 16;
        int coln = cbase + lc;                 // this lane's train column
        int colc = (coln < N) ? coln : (N - 1);
        const float* trow = t + (size_t)colc * KD;

        v8f c0 = {0.f,0.f,0.f,0.f,0.f,0.f,0.f,0.f};   // rows 0..15 of tile
        v8f c1 = {0.f,0.f,0.f,0.f,0.f,0.f,0.f,0.f};   // rows 16..31 of tile

#if __has_builtin(__builtin_amdgcn_wmma_f32_16x16x4_f32)
        #pragma unroll
        for (int kk = 0; kk < KD / 4; ++kk) {
            const int ko = kk * 4 + hi * 2;   // this lane's K pair
            v2f b  = *(const v2f*)(trow + ko);                       // B[k][n]
            v2f a0 = *(const v2f*)(&qlds[lc * QP + ko]);             // A rows 0..15
            v2f a1 = *(const v2f*)(&qlds[(lc + 16) * QP + ko]);      // A rows 16..31
            c0 = __builtin_amdgcn_wmma_f32_16x16x4_f32(false, a0, false, b,
                                                       (short)0, c0, false, false);
            c1 = __builtin_amdgcn_wmma_f32_16x16x4_f32(false, a1, false, b,
                                                       (short)0, c1, false, false);
        }
#else
        // scalar fallback producing the identical lane layout
        for (int e = 0; e < 8; ++e) {
            const int m0 = e + hi * 8;
            float s0 = 0.f, s1 = 0.f;
            for (int k = 0; k < KD; ++k) {
                float tv = trow[k];
                s0 += qlds[m0 * QP + k] * tv;
                s1 += qlds[(m0 + 16) * QP + k] * tv;
            }
            c0[e] = s0; c1[e] = s1;
        }
#endif
        const float tnv = tn[colc];
        // C layout: lane ln, element e -> (M = e + hi*8, Ncol = lc)
        #pragma unroll
        for (int e = 0; e < 8; ++e) {
            sc[wv][(e + hi * 8) * 17 + lc]      = c0[e] + tnv;
            sc[wv][(16 + e + hi * 8) * 17 + lc] = c1[e] + tnv;
        }
        // wave-internal LDS store->load ordering (DS ops are in-order per wave)
        asm volatile("s_wait_dscnt 0" ::: "memory");

        // streaming top-k: lane ln owns row ln; scan the 16 fresh scores
        #pragma unroll 1
        for (int j = 0; j < 16; ++j) {
            const int col = cbase + j;
            const float v = sc[wv][ln * 17 + j];
            if (col < N && v < curMax) {
                int   ms = 0;  float mv = topv[wv][ln][0];
                #pragma unroll
                for (int s = 1; s < TOPK; ++s) {
                    float tv = topv[wv][ln][s];
                    if (tv > mv) { mv = tv; ms = s; }
                }
                topv[wv][ln][ms] = v;
                topi[wv][ln][ms] = col;
                float nm = topv[wv][ln][0];
                #pragma unroll
                for (int s = 1; s < TOPK; ++s) nm = fmaxf(nm, topv[wv][ln][s]);
                curMax = nm;
            }
        }
        asm volatile("s_wait_dscnt 0" ::: "memory");
    }
    __syncthreads();

    // merge 4 waves' top-k per row -> slice top-k; threads 0..31 each own a row
    if (tid < MT) {
        const int r = tid;
        const int qrow = qbase + r;
        if (qrow < B) {
            float* oV = candV + ((size_t)qrow * NS + blockIdx.y) * TOPK;
            int*   oI = candI + ((size_t)qrow * NS + blockIdx.y) * TOPK;
            for (int s = 0; s < TOPK; ++s) {
                float best = SENT; int bw = 0, bs = 0;
                for (int w = 0; w < WAVES; ++w)
                    for (int u = 0; u < TOPK; ++u) {
                        float v = topv[w][r][u];
                        if (v < best) { best = v; bw = w; bs = u; }
                    }
                oV[s] = best;
                oI[s] = topi[bw][r][bs];
                topv[bw][r][bs] = SENT;   // extract
            }
        }
    }
}

// ---------------- kernel 2: merge slices, convert to nonconformity ----------------
__global__ void knn_phase2(const float* __restrict__ candV, const int* __restrict__ candI,
                           const float* __restrict__ qn, const int* __restrict__ lb,
                           float* __restrict__ out, int B, int NS, int L) {
    int qi = blockIdx.x * blockDim.x + threadIdx.x;
    if (qi >= B) return;
    const float* cv = candV + (size_t)qi * NS * TOPK;
    const int*   ci = candI + (size_t)qi * NS * TOPK;

    float vals[TOPK]; int idxs[TOPK];
    for (int s = 0; s < TOPK; ++s) { vals[s] = SENT; idxs[s] = 0; }
    float curMax = SENT;
    const int total = NS * TOPK;
    for (int i = 0; i < total; ++i) {
        float v = cv[i];
        if (v < curMax) {
            int ms = 0; float mv = vals[0];
            for (int s = 1; s < TOPK; ++s) if (vals[s] > mv) { mv = vals[s]; ms = s; }
            vals[ms] = v; idxs[ms] = ci[i];
            mv = vals[0];
            for (int s = 1; s < TOPK; ++s) mv = fmaxf(mv, vals[s]);
            curMax = mv;
        }
    }
    const float qnv = qn[qi];
    float tot = 0.f;
    float perLab[16];
    for (int l = 0; l < 16; ++l) perLab[l] = 0.f;
    for (int s = 0; s < TOPK; ++s) {
        if (vals[s] > 1.0e37f) continue;
        float d2  = vals[s] + qnv;                    // ||x||^2 - 2x.t + ||t||^2
        float inv = 1.0f / sqrtf(fmaxf(d2, 1e-12f));  // 1/dist (matches reference)
        tot += inv;
        int lab = lb[idxs[s]];
        if ((unsigned)lab < 16u) perLab[lab] += inv;
    }
    for (int l = 0; l < L; ++l) out[(size_t)qi * L + l] = tot - perLab[l];
}

// ---------------- host launcher ----------------
extern "C" void kernel_launch(void* const* d_in, const int* in_sizes, int n_in,
                              void* d_out, int out_size, void* d_ws, size_t ws_size,
                              hipStream_t stream) {
    const float* q   = (const float*)d_in[0];   // [B, D] fp32
    const float* t   = (const float*)d_in[1];   // [N, D] fp32
    const int*   lab = (const int*)d_in[2];     // [N] int32
    const int N = in_sizes[2];
    const int D = in_sizes[1] / N;              // 128
    const int B = in_sizes[0] / D;              // 1024
    const int L = in_sizes[3];                  // 10
    const int NS = (N + SLICE - 1) / SLICE;     // 25 slices

    float* tn    = (float*)d_ws;                       // [N]
    float* qn    = tn + N;                             // [B]
    float* candV = qn + B;                             // [B, NS, 32]
    int*   candI = (int*)(candV + (size_t)B * NS * TOPK);

    const int tot = N + B;
    knn_norms<<<(tot + 255) / 256, 256, 0, stream>>>(q, t, tn, qn, B, N, D);

    dim3 g1((B + MT - 1) / MT, NS);
    knn_phase1<<<g1, WAVES * 32, 0, stream>>>(q, t, tn, candV, candI, B, N, NS);

    knn_phase2<<<(B + 127) / 128, 128, 0, stream>>>(candV, candI, qn, lab,
                                                    (float*)d_out, B, NS, L);
}